// HashGrid1D_19645180412074
// MI455X (gfx1250) — compile-verified
//
#include <hip/hip_runtime.h>
#include <stdint.h>
#include <stddef.h>

// ---------------------------------------------------------------------------
// 1D multiresolution hash-grid encode for MI455X (gfx1250).
//
// Strategy:
//   * Levels 0..10 active table slices (262,096 B total) are DMA'd into LDS
//     with CDNA5 TENSOR_LOAD_TO_LDS (one 2D-tile descriptor per level:
//     2 floats wide, stride 24 floats, rows = res_l + 1).
//   * Per-point gathers for those levels are served by LDS (ds_load_b64),
//     avoiding ~88M divergent 8B L2 requests.
//   * Level 11 wraps the full 16384-entry table; its slice (128 KB) stays
//     L2-resident and is gathered from global memory.
//   * Output: 24 f32 per point, written as 6x b128 stores (rows are 96 B,
//     16 B aligned).
// ---------------------------------------------------------------------------

#define HG_LEVELS     12
#define HG_LDS_LEVELS 11
#define HG_HASH       16384
#define HG_ROWF       24          // floats per table row (12 levels * 2 feat)

typedef __attribute__((ext_vector_type(4))) unsigned int u32x4;
typedef __attribute__((ext_vector_type(8))) unsigned int u32x8;

// rows staged for level l: res_l + 1 entries, capped at HASH_SIZE (level 10)
constexpr int hg_rows(int l) { return l < 10 ? (16 << l) + 1 : HG_HASH; }
// byte offset of level l's slice inside LDS (entries are float2 = 8 bytes)
constexpr int hg_off(int l) {
    int o = 0;
    for (int k = 0; k < l; ++k) o += hg_rows(k) * 8;
    return o;
}
// total staged bytes: 262,096  (fits 320 KB LDS)

// Issue one TDM 2D-tile load: rows x 2 floats, row stride 24 floats,
// global -> LDS.  Descriptor layout per CDNA5 ISA.
__device__ __forceinline__ void hg_tdm_load(uint32_t lds_byte_addr,
                                            const float* gsrc,
                                            uint32_t rows) {
    const uint64_t ga = (uint64_t)(uintptr_t)gsrc;

    u32x4 g0;
    g0.x = 1u;                                    // count=1, user-mode, no gather
    g0.y = lds_byte_addr;                         // lds_addr
    g0.z = (uint32_t)ga;                          // global_addr[31:0]
    g0.w = ((uint32_t)(ga >> 32) & 0x01FFFFFFu)   // global_addr[56:32]
         | (2u << 30);                            // type = 2 ("image")

    u32x8 g1;
    g1[0] = 2u << 16;                             // data_size = 4 bytes; mask=0
    g1[1] = 2u << 16;                             // tensor_dim0 = 2 elems
    g1[2] = (rows & 0xFFFFu) << 16;               // tensor_dim1[15:0]
    g1[3] = ((rows >> 16) & 0xFFFFu)              // tensor_dim1[31:16]
          | (2u << 16);                           // tile_dim0 = 2 elems
    g1[4] = rows & 0xFFFFu;                       // tile_dim1 = rows, tile_dim2 = 0
    g1[5] = (uint32_t)HG_ROWF;                    // tensor_dim0_stride = 24 elems
    g1[6] = 0u;                                   // stride hi / dim1_stride lo
    g1[7] = 0u;                                   // dim1_stride hi

    asm volatile("tensor_load_to_lds %0, %1" :: "s"(g0), "s"(g1) : "memory");
}

__device__ __forceinline__ void hg_wait_tensor0() {
#if defined(__has_builtin)
#if __has_builtin(__builtin_amdgcn_s_wait_tensorcnt)
    __builtin_amdgcn_s_wait_tensorcnt(0);
    return;
#endif
#endif
    asm volatile("s_wait_tensorcnt 0x0" ::: "memory");
}

__global__ __launch_bounds__(1024)
void hashgrid1d_kernel(const float* __restrict__ x,
                       const float* __restrict__ table,
                       float* __restrict__ out,
                       int n) {
    extern __shared__ unsigned char smem[];

    // --- Stage levels 0..10 into LDS with the Tensor Data Mover (wave 0) ---
    if (threadIdx.x < 32) {
        const uint32_t lds0 = (uint32_t)(uintptr_t)smem;  // low 32b = LDS offset
#pragma unroll
        for (int l = 0; l < HG_LDS_LEVELS; ++l) {
            hg_tdm_load(lds0 + (uint32_t)hg_off(l), table + 2 * l,
                        (uint32_t)hg_rows(l));
        }
        hg_wait_tensor0();
    }
    __syncthreads();

    const int stride = (int)(gridDim.x * blockDim.x);
    for (int i = (int)(blockIdx.x * blockDim.x + threadIdx.x); i < n; i += stride) {
        const float xv = x[i];
        const float xc = fminf(fmaxf(xv, 0.0f), 1.0f);

        float o[HG_LEVELS * 2];

        // Levels 0..10: gather from LDS
#pragma unroll
        for (int l = 0; l < HG_LDS_LEVELS; ++l) {
            const float res = (float)(16 << l);
            const float pos = xc * res;
            const float fi  = floorf(pos);
            const float w   = pos - fi;
            const int   i0  = (int)fi;
            int a, b;
            if (l == 10) {                       // res == HASH_SIZE: wraps
                a = i0 & (HG_HASH - 1);
                b = (i0 + 1) & (HG_HASH - 1);
            } else {                             // res+1 entries staged: no wrap
                a = i0;
                b = i0 + 1;
            }
            const float2* t = (const float2*)(smem + hg_off(l));
            const float2 e0 = t[a];
            const float2 e1 = t[b];
            o[2 * l + 0] = (1.0f - w) * e0.x + w * e1.x;
            o[2 * l + 1] = (1.0f - w) * e0.y + w * e1.y;
        }

        // Level 11 (res 32768, wraps full table): gather from global (L2-hot)
        {
            const float pos = xc * 32768.0f;
            const float fi  = floorf(pos);
            const float w   = pos - fi;
            const int   i0  = (int)fi;
            const int a = i0 & (HG_HASH - 1);
            const int b = (i0 + 1) & (HG_HASH - 1);
            const float2 e0 = *(const float2*)(table + (size_t)a * HG_ROWF + 22);
            const float2 e1 = *(const float2*)(table + (size_t)b * HG_ROWF + 22);
            o[22] = (1.0f - w) * e0.x + w * e1.x;
            o[23] = (1.0f - w) * e0.y + w * e1.y;
        }

        // 96 B per point, 16 B aligned -> 6x global_store_b128
        float4* vo = (float4*)(out + (size_t)i * (HG_LEVELS * 2));
#pragma unroll
        for (int k = 0; k < 6; ++k) {
            vo[k] = make_float4(o[4 * k + 0], o[4 * k + 1],
                                o[4 * k + 2], o[4 * k + 3]);
        }
    }
}

extern "C" void kernel_launch(void* const* d_in, const int* in_sizes, int n_in,
                              void* d_out, int out_size, void* d_ws, size_t ws_size,
                              hipStream_t stream) {
    const float* x     = (const float*)d_in[0];
    const float* table = (const float*)d_in[1];
    float*       out   = (float*)d_out;
    const int n = in_sizes[0];

    constexpr size_t LDS_BYTES = 262144;  // staged: 262,096 B (<= 320 KB WGP LDS)
    dim3 grid(512), block(1024);
    hipLaunchKernelGGL(hashgrid1d_kernel, grid, block, LDS_BYTES, stream,
                       x, table, out, n);
}